// CellListNLPyTorch_57913339019803
// MI455X (gfx1250) — compile-verified
//
#include <hip/hip_runtime.h>
#include <hip/hip_bf16.h>
#include <stdint.h>

#define NATOMS 4096
#define RC2f   25.0f
#define JT     256   // j-tile width per block (== blockDim.x)
#define IT     16    // i rows per block

typedef __attribute__((ext_vector_type(2))) float v2f;
typedef __attribute__((ext_vector_type(8))) float v8f;

typedef __attribute__((address_space(1))) int gint_t;
typedef __attribute__((address_space(3))) int lint_t;

// ---------------------------------------------------------------------------
// Kernel 1: frac = wrap(coord @ inverse(cell)) using V_WMMA_F32_16X16X4_F32.
// One wave computes frac for 16 atoms: A = 16x4 (atom rows, K = xyz + pad),
// B = 4x16 (rows = K component of H^-1, cols N = frac component, N>=3 pad 0).
// Full fp32 matrix path -> numerically equivalent to the reference matmul.
// ---------------------------------------------------------------------------
__global__ __launch_bounds__(128) void frac_wmma_kernel(
    const float* __restrict__ coord, const float* __restrict__ cell,
    float* __restrict__ frac) {
  // 3x3 inverse via adjugate (uniform, computed per-thread; trivially cheap)
  float c00 = cell[0], c01 = cell[1], c02 = cell[2];
  float c10 = cell[3], c11 = cell[4], c12 = cell[5];
  float c20 = cell[6], c21 = cell[7], c22 = cell[8];
  float i00 = c11 * c22 - c12 * c21;
  float i01 = c02 * c21 - c01 * c22;
  float i02 = c01 * c12 - c02 * c11;
  float i10 = c12 * c20 - c10 * c22;
  float i11 = c00 * c22 - c02 * c20;
  float i12 = c02 * c10 - c00 * c12;
  float i20 = c10 * c21 - c11 * c20;
  float i21 = c01 * c20 - c00 * c21;
  float i22 = c00 * c11 - c01 * c10;
  float rdet = 1.0f / (c00 * i00 + c01 * i10 + c02 * i20);
  float hinv[3][3] = {{i00 * rdet, i01 * rdet, i02 * rdet},
                      {i10 * rdet, i11 * rdet, i12 * rdet},
                      {i20 * rdet, i21 * rdet, i22 * rdet}};

  const int lane = threadIdx.x & 31;
  const int wave = threadIdx.x >> 5;
  const int base = (blockIdx.x * 4 + wave) * 16;  // first atom of this wave's tile
  const int m    = lane & 15;
  const bool hi  = lane >= 16;

  // A (16x4): lanes 0-15 hold K=0,1 in v0,v1 for row M=lane;
  //           lanes 16-31 hold K=2,3 for row M=lane-16 (K=3 padded 0).
  v2f a;
  if (!hi) { a.x = coord[(base + m) * 3 + 0]; a.y = coord[(base + m) * 3 + 1]; }
  else     { a.x = coord[(base + m) * 3 + 2]; a.y = 0.0f; }

  // B (4x16): v0: lanes 0-15 = row K=0 (N=lane), lanes 16-31 = row K=2;
  //           v1: lanes 0-15 = row K=1, lanes 16-31 = row K=3 (pad 0).
  float bk0 = (m < 3) ? hinv[0][m] : 0.0f;
  float bk1 = (m < 3) ? hinv[1][m] : 0.0f;
  float bk2 = (m < 3) ? hinv[2][m] : 0.0f;
  v2f b;
  if (!hi) { b.x = bk0; b.y = bk1; }
  else     { b.x = bk2; b.y = 0.0f; }

  v8f acc = {};
  // 8 args: (neg_a, A, neg_b, B, c_mod, C, reuse_a, reuse_b)
  acc = __builtin_amdgcn_wmma_f32_16x16x4_f32(false, a, false, b, (short)0,
                                              acc, false, false);

  // D layout: VGPR r, lanes 0-15 -> (M=r, N=lane); lanes 16-31 -> (M=r+8).
  // Only N<3 columns are real frac components; wrap into [0,1) and store.
  if (m < 3) {
#pragma unroll
    for (int r = 0; r < 8; ++r) {
      int   M = r + (hi ? 8 : 0);
      float f = acc[r];
      f = f - floorf(f);
      frac[(base + M) * 3 + m] = f;
    }
  }
}

// ---------------------------------------------------------------------------
// Kernel 2: pairwise MIC table. Store-bandwidth bound (256 MB of output).
// j-tile of frac staged into LDS via gfx1250 async global->LDS loads;
// outputs streamed with non-temporal stores to keep L2 clean.
// ---------------------------------------------------------------------------
__global__ __launch_bounds__(256) void pair_mic_kernel(
    const float* __restrict__ frac, const float* __restrict__ cell,
    float* __restrict__ d2o, int* __restrict__ so) {
  __shared__ float fr[JT * 3];

  const int t  = threadIdx.x;
  const int j0 = blockIdx.x * JT;
  const int i0 = blockIdx.y * IT;

  const float* gsrc = frac + (size_t)j0 * 3;  // 768 contiguous floats

#if __has_builtin(__builtin_amdgcn_global_load_async_to_lds_b32)
#pragma unroll
  for (int k = 0; k < 3; ++k) {
    __builtin_amdgcn_global_load_async_to_lds_b32(
        (gint_t*)(gsrc + t + k * JT),
        (lint_t*)(&fr[t + k * JT]),
        0, 0);
  }
#if __has_builtin(__builtin_amdgcn_s_wait_asynccnt)
  __builtin_amdgcn_s_wait_asynccnt(0);
#else
  asm volatile("s_wait_asynccnt 0" ::: "memory");
#endif
#else
#pragma unroll
  for (int k = 0; k < 3; ++k) fr[t + k * JT] = gsrc[t + k * JT];
#endif
  __syncthreads();

  // uniform cell (scalar-register resident)
  float h00 = cell[0], h01 = cell[1], h02 = cell[2];
  float h10 = cell[3], h11 = cell[4], h12 = cell[5];
  float h20 = cell[6], h21 = cell[7], h22 = cell[8];

  const float fj0 = fr[t * 3 + 0];
  const float fj1 = fr[t * 3 + 1];
  const float fj2 = fr[t * 3 + 2];
  const size_t jg = (size_t)(j0 + t);

#pragma unroll 4
  for (int it = 0; it < IT; ++it) {
    const int i = i0 + it;
    // uniform address -> scalar loads, serviced from WGP$/L2
    float fi0 = frac[i * 3 + 0];
    float fi1 = frac[i * 3 + 1];
    float fi2 = frac[i * 3 + 2];

    float df0 = fj0 - fi0, df1 = fj1 - fi1, df2 = fj2 - fi2;
    float r0 = rintf(df0), r1 = rintf(df1), r2 = rintf(df2);  // v_rndne_f32
    float u0 = df0 - r0, u1 = df1 - r1, u2 = df2 - r2;

    float x = u0 * h00 + u1 * h10 + u2 * h20;
    float y = u0 * h01 + u1 * h11 + u2 * h21;
    float z = u0 * h02 + u1 * h12 + u2 * h22;
    float dist2 = x * x + y * y + z * z;

    bool keep = (dist2 > 0.0f) && (dist2 < RC2f);
    float dout = keep ? dist2 : 0.0f;
    int s0 = keep ? -(int)r0 : 0;
    int s1 = keep ? -(int)r1 : 0;
    int s2 = keep ? -(int)r2 : 0;

    size_t off = (size_t)i * NATOMS + jg;
    __builtin_nontemporal_store(dout, d2o + off);
    __builtin_nontemporal_store(s0, so + off * 3 + 0);
    __builtin_nontemporal_store(s1, so + off * 3 + 1);
    __builtin_nontemporal_store(s2, so + off * 3 + 2);
  }
}

// ---------------------------------------------------------------------------
extern "C" void kernel_launch(void* const* d_in, const int* in_sizes, int n_in,
                              void* d_out, int out_size, void* d_ws,
                              size_t ws_size, hipStream_t stream) {
  const float* coord = (const float*)d_in[0];  // [4096,3] f32
  const float* cell  = (const float*)d_in[1];  // [3,3]    f32

  float* frac = (float*)d_ws;  // 4096*3 floats = 48 KB scratch

  float* d2o = (float*)d_out;                           // [n,n] f32
  int*   so  = (int*)d_out + (size_t)NATOMS * NATOMS;   // [n,n,3] i32

  // 64 blocks x 4 waves x 16 atoms = 4096 atoms
  frac_wmma_kernel<<<NATOMS / 64, 128, 0, stream>>>(coord, cell, frac);

  dim3 grid(NATOMS / JT, NATOMS / IT);  // (16, 256)
  pair_mic_kernel<<<grid, JT, 0, stream>>>(frac, cell, d2o, so);
}